// MessagePass_50173807952912
// MI455X (gfx1250) — compile-verified
//
#include <hip/hip_runtime.h>
#include <hip/hip_bf16.h>

typedef __attribute__((ext_vector_type(2))) float v2f;
typedef __attribute__((ext_vector_type(4))) float v4f;
typedef __attribute__((ext_vector_type(8))) float v8f;

#define BB 128
#define LL 1024
#define CC 16
#define NN 65536

// ---------------------------------------------------------------------------
// 1) starts[b] = lower_bound(coeffs_batch, b)   (coeffs_batch is sorted int64)
// ---------------------------------------------------------------------------
__global__ void mp_starts_kernel(const long long* __restrict__ batch,
                                 int* __restrict__ starts) {
  int b = blockIdx.x * blockDim.x + threadIdx.x;
  if (b >= BB) return;
  int lo = 0, hi = NN;
  while (lo < hi) {
    int mid = (lo + hi) >> 1;
    if (batch[mid] < (long long)b) lo = mid + 1; else hi = mid;
  }
  starts[b] = lo;
}

// ---------------------------------------------------------------------------
// 2) zero dense scratch (grid-stride float4)
// ---------------------------------------------------------------------------
__global__ void mp_zero_kernel(v4f* __restrict__ p, int n4) {
  int i = blockIdx.x * blockDim.x + threadIdx.x;
  int stride = gridDim.x * blockDim.x;
  v4f z = {0.f, 0.f, 0.f, 0.f};
  for (; i < n4; i += stride) p[i] = z;
}

// ---------------------------------------------------------------------------
// 3) scatter x[N,16] -> dense[B,L,16]
// ---------------------------------------------------------------------------
__global__ void mp_scatter_kernel(const float* __restrict__ x,
                                  const long long* __restrict__ batch,
                                  const int* __restrict__ starts,
                                  float* __restrict__ dense) {
  int n = blockIdx.x * blockDim.x + threadIdx.x;
  if (n >= NN) return;
  int b = (int)batch[n];
  int p = n - starts[b];
  if (p > LL - 1) p = LL - 1;           // jax clip semantics
  const v4f* src = (const v4f*)(x + (size_t)n * CC);
  v4f* dst = (v4f*)(dense + ((size_t)b * LL + p) * CC);
#pragma unroll
  for (int j = 0; j < 4; ++j) dst[j] = src[j];
}

// ---------------------------------------------------------------------------
// 4) batched GEMM: y[b] = M[b] (LxL) @ dense[b] (LxC) via v_wmma_f32_16x16x4
//    grid = (8, B); block = 256 (8 waves); wave w owns rows [tile*128+w*16, +16)
//    A stream (512 MB, read once) uses non-temporal loads to keep it from
//    churning the 192 MB L2; dense/y (reused) stay regular-temporal.
// ---------------------------------------------------------------------------
__global__ __launch_bounds__(256) void mp_gemm_kernel(
    const float* __restrict__ M, const float* __restrict__ dense,
    float* __restrict__ y) {
  __shared__ float sB[LL * CC];         // 64 KB: dense[b], k-major rows of 16

  const int b    = blockIdx.y;
  const int tile = blockIdx.x;          // 0..7 -> 128 output rows each
  const int t    = threadIdx.x;

  // cooperative LDS fill: 4096 float4 / 256 threads = 16 each
  {
    const v4f* g = (const v4f*)(dense + (size_t)b * LL * CC);
    v4f* s = (v4f*)sB;
#pragma unroll
    for (int j = 0; j < 16; ++j) s[t + j * 256] = g[t + j * 256];
  }
  __syncthreads();

  const int wave = t >> 5;
  const int lane = t & 31;
  const int hi   = lane >> 4;           // 0: K=0,1   1: K=2,3
  const int ln   = lane & 15;

  const int row0 = tile * 128 + wave * 16;       // tile base row
  const int arow = row0 + ln;                    // A-matrix row for this lane
  const float* Arow = M + ((size_t)b * LL + arow) * LL + 2 * hi;
  const float* Bcol = sB + 2 * hi * CC + ln;     // sB[(k+2hi)*16 + ln]

  v8f acc0 = {};
  v8f acc1 = {};

#pragma unroll 4
  for (int k0 = 0; k0 < LL; k0 += 8) {
    // ---- WMMA #0: K = k0..k0+3 ----
    v2f a0 = __builtin_nontemporal_load((const v2f*)(Arow + k0));
    v2f b0;
    b0.x = Bcol[(size_t)k0 * CC];                // conflict-free ds loads
    b0.y = Bcol[(size_t)(k0 + 1) * CC];
    acc0 = __builtin_amdgcn_wmma_f32_16x16x4_f32(
        false, a0, false, b0, (short)0, acc0, false, false);

    // ---- WMMA #1: K = k0+4..k0+7 (independent accumulator) ----
    v2f a1 = __builtin_nontemporal_load((const v2f*)(Arow + k0 + 4));
    v2f b1;
    b1.x = Bcol[(size_t)(k0 + 4) * CC];
    b1.y = Bcol[(size_t)(k0 + 5) * CC];
    acc1 = __builtin_amdgcn_wmma_f32_16x16x4_f32(
        false, a1, false, b1, (short)0, acc1, false, false);
  }

  v8f acc = acc0 + acc1;

  // C/D layout: VGPR v, lane -> row = v + hi*8, col = ln
  float* yrow = y + ((size_t)b * LL + row0 + hi * 8) * CC + ln;
#pragma unroll
  for (int v = 0; v < 8; ++v) yrow[(size_t)v * CC] = acc[v];
}

// ---------------------------------------------------------------------------
// 5) gather y[B,L,16] -> out[N,16]
// ---------------------------------------------------------------------------
__global__ void mp_gather_kernel(const float* __restrict__ y,
                                 const long long* __restrict__ batch,
                                 const int* __restrict__ starts,
                                 float* __restrict__ out) {
  int n = blockIdx.x * blockDim.x + threadIdx.x;
  if (n >= NN) return;
  int b = (int)batch[n];
  int p = n - starts[b];
  if (p > LL - 1) p = LL - 1;
  const v4f* src = (const v4f*)(y + ((size_t)b * LL + p) * CC);
  v4f* dst = (v4f*)(out + (size_t)n * CC);
#pragma unroll
  for (int j = 0; j < 4; ++j) dst[j] = src[j];
}

// ---------------------------------------------------------------------------
extern "C" void kernel_launch(void* const* d_in, const int* in_sizes, int n_in,
                              void* d_out, int out_size, void* d_ws,
                              size_t ws_size, hipStream_t stream) {
  const float*     x     = (const float*)d_in[0];
  const float*     M     = (const float*)d_in[1];
  const long long* batch = (const long long*)d_in[2];
  float*           out   = (float*)d_out;

  const size_t denseBytes = (size_t)BB * LL * CC * sizeof(float);  // 8 MB
  char* ws     = (char*)d_ws;
  int*   starts = (int*)ws;                                   // 512 B slot
  float* dense  = (float*)(ws + 1024);
  float* y      = (float*)(ws + 1024 + denseBytes);

  mp_starts_kernel<<<1, 128, 0, stream>>>(batch, starts);

  const int n4 = BB * LL * CC / 4;
  mp_zero_kernel<<<512, 256, 0, stream>>>((v4f*)dense, n4);

  mp_scatter_kernel<<<NN / 256, 256, 0, stream>>>(x, batch, starts, dense);

  dim3 grid(LL / 128, BB);   // (8, 128)
  mp_gemm_kernel<<<grid, 256, 0, stream>>>(M, dense, y);

  mp_gather_kernel<<<NN / 256, 256, 0, stream>>>(y, batch, starts, out);
}